// HeteroGCN_30709016166547
// MI455X (gfx1250) — compile-verified
//
#include <hip/hip_runtime.h>
#include <hip/hip_bf16.h>

// CDNA5 / gfx1250: wave32, WMMA 16x16 tiles.
typedef __attribute__((ext_vector_type(2))) float v2f;
typedef __attribute__((ext_vector_type(8))) float v8f;

#define WAVE 32

// ---------------------------------------------------------------------------
// Dual GEMM: Ha = act(A) @ Wa ; Hb = act(A) @ Wb      (A: [n x 64], W: 64x64)
// act = relu if RELU else identity (compile-time -> no per-element selects).
//
// Both relations share the same A matrix, so fusing halves the dominant
// A-stream traffic (25.6 MB vs two 16 KB weight matrices). Weights are staged
// once per block into LDS (2 x 16 KB out of 320 KB/WGP) so the K-loop's
// B-fragment reads ride the DS pipe while VMEM streams A.
//
// One wave per 16x16 output tile, K-loop of 16 x V_WMMA_F32_16X16X4_F32
// per output (32 WMMA/wave). ISA layouts (CDNA5 ISA §7.12.2):
//   A 16x4 f32 : lane L -> M = L%16 ; VGPR0/1 hold K = 2*(L/16) + {0,1}
//   B 4x16 f32 : lane L -> N = L%16 ; VGPR0/1 hold K = 2*(L/16) + {0,1}
//   C/D 16x16  : lane L -> N = L%16 ; VGPR r holds M = r + 8*(L/16)
// EXEC must be all-1s at the WMMA: bounds handled by clamped loads +
// predicated stores only (no divergence around the WMMA loop).
// ---------------------------------------------------------------------------
template <int RELU>
__global__ void gemm64_dual_wmma(const float* __restrict__ A,
                                 const float* __restrict__ Wa,
                                 const float* __restrict__ Wb,
                                 float* __restrict__ Ha,
                                 float* __restrict__ Hb,
                                 int n)
{
    __shared__ float Wl[2][64 * 64];          // 2 x 16 KB

    // Cooperative stage of both weight matrices (float4, fully coalesced).
    for (int i = threadIdx.x; i < 64 * 16; i += blockDim.x) {
        ((float4*)Wl[0])[i] = ((const float4*)Wa)[i];
        ((float4*)Wl[1])[i] = ((const float4*)Wb)[i];
    }
    __syncthreads();

    const int wavesPerBlock = blockDim.x / WAVE;
    const int waveId = blockIdx.x * wavesPerBlock + (int)(threadIdx.x / WAVE);
    const int lane   = (int)(threadIdx.x & (WAVE - 1));
    const int rowTiles = (n + 15) >> 4;
    const int rowTile  = waveId >> 2;          // 64 cols -> 4 col tiles
    const int colTile  = waveId & 3;
    if (rowTile >= rowTiles) return;           // uniform per-wave exit

    const int half = lane >> 4;                // 0: K {0,1}, 1: K {2,3}
    const int l16  = lane & 15;
    const int row  = rowTile * 16 + l16;       // A-operand row (M)
    const int rowc = (row < n) ? row : (n - 1);
    const int col  = colTile * 16 + l16;       // B/C/D column (N)

    const float* __restrict__ arow = A + (size_t)rowc * 64;

    v8f ca = {0.f, 0.f, 0.f, 0.f, 0.f, 0.f, 0.f, 0.f};
    v8f cb = {0.f, 0.f, 0.f, 0.f, 0.f, 0.f, 0.f, 0.f};

#pragma unroll
    for (int k = 0; k < 64; k += 4) {
        const int kb = k + half * 2;
        float a0 = arow[kb + 0];
        float a1 = arow[kb + 1];
        if (RELU) { a0 = fmaxf(a0, 0.f); a1 = fmaxf(a1, 0.f); }
        v2f a; a.x = a0; a.y = a1;

        v2f b0;
        b0.x = Wl[0][(kb + 0) * 64 + col];
        b0.y = Wl[0][(kb + 1) * 64 + col];
        ca = __builtin_amdgcn_wmma_f32_16x16x4_f32(
                 false, a, false, b0, (short)0, ca, false, false);

        v2f b1;
        b1.x = Wl[1][(kb + 0) * 64 + col];
        b1.y = Wl[1][(kb + 1) * 64 + col];
        cb = __builtin_amdgcn_wmma_f32_16x16x4_f32(
                 false, a, false, b1, (short)0, cb, false, false);
    }

#pragma unroll
    for (int r = 0; r < 8; ++r) {
        const int m = rowTile * 16 + r + 8 * half;
        if (m < n) {
            Ha[(size_t)m * 64 + col] = ca[r];
            Hb[(size_t)m * 64 + col] = cb[r];
        }
    }
}

// ---------------------------------------------------------------------------
// Degree handling: deg initialized to 1.0 (the self-loop), then +1 per
// incoming edge via global_atomic_add_f32, then rsqrt in place.
// ---------------------------------------------------------------------------
__global__ void init_deg(float* __restrict__ deg1, float* __restrict__ deg2, int n)
{
    int i = blockIdx.x * blockDim.x + threadIdx.x;
    if (i < n) { deg1[i] = 1.0f; deg2[i] = 1.0f; }
}

__global__ void count_deg(const int* __restrict__ dst, float* __restrict__ deg, int E)
{
    int e = blockIdx.x * blockDim.x + threadIdx.x;
    if (e < E) atomicAdd(&deg[dst[e]], 1.0f);
}

__global__ void make_dinv(float* __restrict__ deg, int n)
{
    int i = blockIdx.x * blockDim.x + threadIdx.x;
    if (i < n) deg[i] = rsqrtf(deg[i]);   // deg >= 1 always
}

// ---------------------------------------------------------------------------
// acc[i][:] = b_a + b_b + dinv1[i]^2 * h1[i][:] + dinv2[i]^2 * h2[i][:]
// (bias + the two self-loop messages; also fully overwrites poisoned output)
// 16 threads per node, float4 per thread.
// ---------------------------------------------------------------------------
__global__ void selfloop_bias(const float* __restrict__ h1,
                              const float* __restrict__ h2,
                              const float* __restrict__ dinv1,
                              const float* __restrict__ dinv2,
                              const float* __restrict__ ba,
                              const float* __restrict__ bb,
                              float* __restrict__ acc, int n)
{
    long long t = (long long)blockIdx.x * blockDim.x + threadIdx.x;
    int i = (int)(t >> 4);
    if (i >= n) return;
    int c4 = (int)(t & 15) * 4;

    float n1 = dinv1[i]; n1 *= n1;
    float n2 = dinv2[i]; n2 *= n2;

    const float4 v1 = *(const float4*)(h1 + (size_t)i * 64 + c4);
    const float4 v2 = *(const float4*)(h2 + (size_t)i * 64 + c4);
    const float4 b1 = *(const float4*)(ba + c4);
    const float4 b2 = *(const float4*)(bb + c4);

    float4 o;
    o.x = b1.x + b2.x + n1 * v1.x + n2 * v2.x;
    o.y = b1.y + b2.y + n1 * v1.y + n2 * v2.y;
    o.z = b1.z + b2.z + n1 * v1.z + n2 * v2.z;
    o.w = b1.w + b2.w + n1 * v1.w + n2 * v2.w;
    *(float4*)(acc + (size_t)i * 64 + c4) = o;
}

// ---------------------------------------------------------------------------
// Edge scatter: msg = h[src] * dinv[src]*dinv[dst] ; acc[dst] += msg.
// 16 threads per edge, float4 gather per thread (coalesced 256B per edge),
// 4x global_atomic_add_f32 scatter. Working set is L2-resident on MI455X
// (192MB), so this is L2-atomic-bandwidth bound. Prefetch the src row of a
// later edge to hide the random-gather latency (global_prefetch_b8).
// ---------------------------------------------------------------------------
__global__ void edge_scatter(const int* __restrict__ src,
                             const int* __restrict__ dst,
                             const float* __restrict__ h,
                             const float* __restrict__ dinv,
                             float* __restrict__ acc, int E)
{
    long long t = (long long)blockIdx.x * blockDim.x + threadIdx.x;
    int e = (int)(t >> 4);
    if (e >= E) return;
    int c4 = (int)(t & 15) * 4;

    if (e + 256 < E) {
        int sn = src[e + 256];
        __builtin_prefetch(h + (size_t)sn * 64 + c4, 0, 1);
    }

    int s = src[e];
    int d = dst[e];
    float norm = dinv[s] * dinv[d];

    float4 v = *(const float4*)(h + (size_t)s * 64 + c4);
    float* ap = acc + (size_t)d * 64 + c4;
    atomicAdd(ap + 0, v.x * norm);
    atomicAdd(ap + 1, v.y * norm);
    atomicAdd(ap + 2, v.z * norm);
    atomicAdd(ap + 3, v.w * norm);
}

// ---------------------------------------------------------------------------
// Host launcher
// ---------------------------------------------------------------------------
extern "C" void kernel_launch(void* const* d_in, const int* in_sizes, int n_in,
                              void* d_out, int out_size, void* d_ws, size_t ws_size,
                              hipStream_t stream)
{
    const float* x    = (const float*)d_in[0];
    const int*   ei1  = (const int*)  d_in[1];   // [2,E] row0=src row1=dst
    const int*   ei2  = (const int*)  d_in[2];
    const float* W1a  = (const float*)d_in[3];
    const float* b1a  = (const float*)d_in[4];
    const float* W1b  = (const float*)d_in[5];
    const float* b1b  = (const float*)d_in[6];
    const float* W2a  = (const float*)d_in[7];
    const float* b2a  = (const float*)d_in[8];
    const float* W2b  = (const float*)d_in[9];
    const float* b2b  = (const float*)d_in[10];
    float* out = (float*)d_out;

    const int N = in_sizes[0] / 64;
    const int E = in_sizes[1] / 2;

    const int* src1 = ei1;     const int* dst1 = ei1 + E;
    const int* src2 = ei2;     const int* dst2 = ei2 + E;

    // Workspace layout (floats): h1 | h2 | acc | dinv1 | dinv2
    float* h1    = (float*)d_ws;
    float* h2    = h1 + (size_t)N * 64;
    float* acc   = h2 + (size_t)N * 64;
    float* dinv1 = acc + (size_t)N * 64;
    float* dinv2 = dinv1 + N;

    const int TB = 256;
    const int rowTiles   = (N + 15) / 16;
    const int totalWaves = rowTiles * 4;                 // 4 col-tiles of 16
    const int gemmBlocks = (totalWaves + (TB / WAVE) - 1) / (TB / WAVE);
    const int nodeBlocks = (N + TB - 1) / TB;
    const int edgeBlocks = (E + TB - 1) / TB;
    const long long nodeV = (long long)N * 16;           // 16 thr/node
    const long long edgeV = (long long)E * 16;           // 16 thr/edge
    const int nodeVBlocks = (int)((nodeV + TB - 1) / TB);
    const int edgeVBlocks = (int)((edgeV + TB - 1) / TB);

    // --- degrees (self-loop folded in as init=1.0) ---
    init_deg<<<nodeBlocks, TB, 0, stream>>>(dinv1, dinv2, N);
    count_deg<<<edgeBlocks, TB, 0, stream>>>(dst1, dinv1, E);
    count_deg<<<edgeBlocks, TB, 0, stream>>>(dst2, dinv2, E);
    make_dinv<<<nodeBlocks, TB, 0, stream>>>(dinv1, N);
    make_dinv<<<nodeBlocks, TB, 0, stream>>>(dinv2, N);

    // --- layer 1: h_r = x @ W1_r (both relations in one fused pass) ---
    gemm64_dual_wmma<0><<<gemmBlocks, TB, 0, stream>>>(x, W1a, W1b, h1, h2, N);
    selfloop_bias<<<nodeVBlocks, TB, 0, stream>>>(h1, h2, dinv1, dinv2, b1a, b1b, acc, N);
    edge_scatter<<<edgeVBlocks, TB, 0, stream>>>(src1, dst1, h1, dinv1, acc, E);
    edge_scatter<<<edgeVBlocks, TB, 0, stream>>>(src2, dst2, h2, dinv2, acc, E);

    // --- layer 2: h_r = relu(acc) @ W2_r (ReLU fused into GEMM A-load) ---
    gemm64_dual_wmma<1><<<gemmBlocks, TB, 0, stream>>>(acc, W2a, W2b, h1, h2, N);
    selfloop_bias<<<nodeVBlocks, TB, 0, stream>>>(h1, h2, dinv1, dinv2, b2a, b2b, out, N);
    edge_scatter<<<edgeVBlocks, TB, 0, stream>>>(src1, dst1, h1, dinv1, out, E);
    edge_scatter<<<edgeVBlocks, TB, 0, stream>>>(src2, dst2, h2, dinv2, out, E);
}